// TopologicalLoss_15865609192042
// MI455X (gfx1250) — compile-verified
//
#include <hip/hip_runtime.h>
#include <hip/hip_bf16.h>
#include <math.h>

// TopologicalLoss for MI455X (gfx1250, wave32).
// Pipeline:
//  1) scale_norm_kernel : s[b,n] = softplus(params)*(1+sigmoid(logits)), ||p||^2
//  2) dist_wmma_kernel  : D[b,i,j] = (i==j)?0 : sqrt(max(ni+nj-2*Gram,0)+EPS)*max(si,sj)
//                         Gram tiles via V_WMMA_F32_16X16X4_F32 (K=3 padded to 4),
//                         one wave = one 16-row strip x 4 16-col tiles (4 WMMAs).
//  3) prim_sort_w2_kernel: per-batch Prim MST (LDS state, rows hit L2),
//                          bitonic sort of deaths & clean diag, W2^2 reduction
//  4) finalize_kernel   : deterministic ordered sum -> 0.1 * mean

typedef __attribute__((ext_vector_type(2))) float v2f;
typedef __attribute__((ext_vector_type(8))) float v8f;

#define TL_EPS 1e-8f
#define TL_WEIGHT 0.1f

__global__ void scale_norm_kernel(const float* __restrict__ points,
                                  const float* __restrict__ params,
                                  const float* __restrict__ logits,
                                  float* __restrict__ scale,
                                  float* __restrict__ norm2,
                                  int total /*B*N*/) {
    int i = blockIdx.x * blockDim.x + threadIdx.x;
    if (i >= total) return;
    float x = params[i];
    float sp = (x > 20.f) ? x : log1pf(expf(x));          // stable softplus
    float lg = logits[i];                                  // [B,N,1] flat
    float sig = 1.f / (1.f + expf(-lg));
    scale[i] = sp * (1.f + sig);
    float px = points[i * 3 + 0];
    float py = points[i * 3 + 1];
    float pz = points[i * 3 + 2];
    norm2[i] = px * px + py * py + pz * pz;
}

// One wave handles rows [i0,i0+16) x columns [j0base, j0base+64): 4 WMMA tiles.
__global__ void dist_wmma_kernel(const float* __restrict__ points,
                                 const float* __restrict__ scale,
                                 const float* __restrict__ norm2,
                                 float* __restrict__ Dm,
                                 int N, int tilesPerDim, long long totalWaves) {
    long long wid = ((long long)blockIdx.x * blockDim.x + threadIdx.x) >> 5;
    if (wid >= totalWaves) return;
    int lane = threadIdx.x & 31;
    int half = lane >> 4;          // 0: K=0,1   1: K=2,3 (zero pad)
    int m    = lane & 15;

    int jGroups = tilesPerDim >> 2;              // 4 j-tiles per wave
    int perBatch = tilesPerDim * jGroups;
    int b   = (int)(wid / perBatch);
    int rem = (int)(wid - (long long)b * perBatch);
    int ti = rem / jGroups;
    int tg = rem - ti * jGroups;
    int i0 = ti << 4;
    int j0base = tg << 6;

    const float* P = points + (size_t)b * N * 3;
    const float* S = scale  + (size_t)b * N;
    const float* Q = norm2  + (size_t)b * N;
    float* Db = Dm + (size_t)b * N * N;

    // A operand: 16x4 tile of P rows i0..i0+15 (K=3 padded with 0).
    // Layout: lanes 0-15 hold K=0,1 (VGPR0,1); lanes 16-31 hold K=2,3.
    int ra = i0 + m;
    v2f a;
    if (half == 0) { a.x = P[ra * 3 + 0]; a.y = P[ra * 3 + 1]; }
    else           { a.x = P[ra * 3 + 2]; a.y = 0.f; }

    // Row-side norms/scales: C/D layout maps VGPR v -> row v (lanes 0-15)
    // or row v+8 (lanes 16-31). Hoisted once per wave.
    int ibase = i0 + (half << 3);
    float ni[8], si[8];
#pragma unroll
    for (int v = 0; v < 8; ++v) {
        ni[v] = Q[ibase + v];
        si[v] = S[ibase + v];
    }

    const size_t rowOff = (size_t)ibase * N + m;   // element (ibase, m)

#pragma unroll
    for (int jt = 0; jt < 4; ++jt) {
        int j0 = j0base + (jt << 4);
        // B operand: 4x16 tile = P^T columns j0..j0+15 (same per-lane packing).
        int ca = j0 + m;
        v2f bb;
        if (half == 0) { bb.x = P[ca * 3 + 0]; bb.y = P[ca * 3 + 1]; }
        else           { bb.x = P[ca * 3 + 2]; bb.y = 0.f; }

        v8f c = {};
        c = __builtin_amdgcn_wmma_f32_16x16x4_f32(
            /*neg_a=*/false, a, /*neg_b=*/false, bb,
            /*c_mod=*/(short)0, c, /*reuse_a=*/false, /*reuse_b=*/false);

        int j = j0 + m;
        float nj = Q[j];
        float sj = S[j];
        size_t base = rowOff + (size_t)j0;   // (ibase, j)
#pragma unroll
        for (int v = 0; v < 8; ++v) {
            float d2 = fmaxf(ni[v] + nj - 2.f * c[v], 0.f) + TL_EPS;
            float eu = __builtin_amdgcn_sqrtf(d2);       // raw v_sqrt_f32
            float val = eu * fmaxf(si[v], sj);           // eu*max(si,sj) symmetrizes
            val = (ibase + v == j) ? 0.f : val;          // zero diagonal (cndmask)
            Db[base + (size_t)v * N] = val;
        }
    }
}

__device__ inline unsigned long long tl_shfl_down_u64(unsigned long long v, int off) {
    unsigned lo = (unsigned)v;
    unsigned hi = (unsigned)(v >> 32);
    lo = __shfl_down(lo, off, 32);
    hi = __shfl_down(hi, off, 32);
    return ((unsigned long long)hi << 32) | lo;
}

// One block per batch; blockDim.x == N (power of two, <= 1024).
__global__ __launch_bounds__(1024)
void prim_sort_w2_kernel(const float* __restrict__ Dm,
                         const float* __restrict__ clean,
                         float* __restrict__ per_sample,
                         int N) {
    const int b = blockIdx.x;
    const int t = threadIdx.x;
    const int lane = t & 31;
    const int wave = t >> 5;

    __shared__ float mind[1024];
    __shared__ int   in_tree[1024];
    __shared__ unsigned long long part[32];
    __shared__ float deaths[1024];
    __shared__ float cleanS[1024];
    __shared__ int   s_j;
    __shared__ float red[1024];

    const float INFV = __int_as_float(0x7f800000);
    const float* Db = Dm + (size_t)b * N * N;

    mind[t]    = Db[t];                 // row 0
    in_tree[t] = (t == 0) ? 1 : 0;
    deaths[t]  = INFV;                  // pad slot N-1 with +inf
    cleanS[t]  = (t < N - 1) ? clean[(size_t)b * (N - 1) + t] : INFV;
    __syncthreads();

    // Prim's algorithm: N-1 sequential argmin + row-relax steps.
    for (int step = 0; step < N - 1; ++step) {
        float val = in_tree[t] ? INFV : mind[t];
        // distances are >= 0 so float bits are order-monotonic; pack (val,idx);
        // min of packed key picks smallest value, ties -> lowest index (jnp.argmin).
        unsigned long long key =
            ((unsigned long long)__float_as_uint(val) << 32) | (unsigned)t;
#pragma unroll
        for (int off = 16; off > 0; off >>= 1) {
            unsigned long long o = tl_shfl_down_u64(key, off);
            if (o < key) key = o;
        }
        if (lane == 0) part[wave] = key;
        __syncthreads();
        if (t < 32) {
            unsigned long long k2 = part[t];
#pragma unroll
            for (int off = 16; off > 0; off >>= 1) {
                unsigned long long o = tl_shfl_down_u64(k2, off);
                if (o < k2) k2 = o;
            }
            if (t == 0) {
                s_j = (int)(k2 & 0xffffffffu);
                deaths[step] = __uint_as_float((unsigned)(k2 >> 32));
            }
        }
        __syncthreads();
        int j = s_j;
        if (t == j) in_tree[t] = 1;
        mind[t] = fminf(mind[t], Db[(size_t)j * N + t]);  // L2-resident row
        __syncthreads();
    }

    // Bitonic sort (ascending) of deaths[] and cleanS[] simultaneously.
    for (int k = 2; k <= N; k <<= 1) {
        for (int jj = k >> 1; jj > 0; jj >>= 1) {
            int ixj = t ^ jj;
            if (ixj > t) {
                bool up = ((t & k) == 0);
                float a0 = deaths[t], a1 = deaths[ixj];
                if ((a0 > a1) == up) { deaths[t] = a1; deaths[ixj] = a0; }
                float c0 = cleanS[t], c1 = cleanS[ixj];
                if ((c0 > c1) == up) { cleanS[t] = c1; cleanS[ixj] = c0; }
            }
            __syncthreads();
        }
    }

    // W2^2 over the N-1 matched pairs; fixed-tree deterministic reduction.
    float dlt = 0.f;
    if (t < N - 1) {
        float dd = deaths[t] - cleanS[t];
        dlt = dd * dd;
    }
    red[t] = dlt;
    __syncthreads();
    for (int s = N >> 1; s > 0; s >>= 1) {
        if (t < s) red[t] += red[t + s];
        __syncthreads();
    }
    if (t == 0) per_sample[b] = red[0];
}

__global__ void finalize_kernel(const float* __restrict__ per_sample,
                                float* __restrict__ out, int B) {
    if (blockIdx.x == 0 && threadIdx.x == 0) {
        float s = 0.f;
        for (int b = 0; b < B; ++b) s += per_sample[b];   // fixed order
        out[0] = TL_WEIGHT * (s / (float)B);
    }
}

extern "C" void kernel_launch(void* const* d_in, const int* in_sizes, int n_in,
                              void* d_out, int out_size, void* d_ws, size_t ws_size,
                              hipStream_t stream) {
    const float* points = (const float*)d_in[0];  // [B,N,3]
    const float* params = (const float*)d_in[1];  // [B,N]
    const float* logits = (const float*)d_in[2];  // [B,N,1]
    const float* clean  = (const float*)d_in[3];  // [B,N-1]

    int BN = in_sizes[1];
    int B  = BN - in_sizes[3];     // B*N - B*(N-1) = B
    int N  = BN / B;               // 1024

    // Workspace layout: D (B*N*N f32), scale, norm2, per_sample
    char* ws = (char*)d_ws;
    size_t szD  = (size_t)B * N * N * sizeof(float);
    size_t offS = (szD + 255) & ~(size_t)255;
    size_t szS  = (size_t)BN * sizeof(float);
    size_t offQ = (offS + szS + 255) & ~(size_t)255;
    size_t offP = (offQ + szS + 255) & ~(size_t)255;
    float* Dm        = (float*)(ws);
    float* scale     = (float*)(ws + offS);
    float* norm2     = (float*)(ws + offQ);
    float* per_samp  = (float*)(ws + offP);

    scale_norm_kernel<<<(BN + 255) / 256, 256, 0, stream>>>(
        points, params, logits, scale, norm2, BN);

    int tilesPerDim = N / 16;                          // 64
    int jGroups = tilesPerDim >> 2;                    // 16
    long long waves = (long long)B * tilesPerDim * jGroups;
    long long totalThreads = waves * 32;
    int blocks = (int)((totalThreads + 255) / 256);
    dist_wmma_kernel<<<blocks, 256, 0, stream>>>(
        points, scale, norm2, Dm, N, tilesPerDim, waves);

    prim_sort_w2_kernel<<<B, N, 0, stream>>>(Dm, clean, per_samp, N);
    finalize_kernel<<<1, 64, 0, stream>>>(per_samp, (float*)d_out, B);
}